// CorticalColumnLinear_62182536512339
// MI455X (gfx1250) — compile-verified
//
#include <hip/hip_runtime.h>

// ---------------------------------------------------------------------------
// out[M,N] = X[M,K] * (W*mask)[N,K]^T   (M=32768, N=K=1024, fp32 io)
// 2:4 structured sparsity along K (row-invariant) => V_SWMMAC_F32_16X16X64_BF16
//   A = compressed weight (16x64 expanded, stored 16x32 bf16)
//   B = X^T tile (dense 64x16, column-major), f32 accumulate, D = out^T tile.
// Memory-bound: ~260 MB @ 23.3 TB/s => ~11 us floor. W pre-compressed to bf16
// once into d_ws; per-tile W staging is GLOBAL_LOAD_ASYNC_TO_LDS_B128
// (ASYNCcnt) double-buffered against swmmac; X converts f32->bf16 via native
// bf16 fptrunc. Epilogue transposes D via LDS for coalesced stores.
// ---------------------------------------------------------------------------

typedef __bf16 v32bf __attribute__((ext_vector_type(32)));
typedef __bf16 v16bf __attribute__((ext_vector_type(16)));
typedef __bf16 v8bf  __attribute__((ext_vector_type(8)));
typedef __bf16 v2bf  __attribute__((ext_vector_type(2)));
typedef float  v8f   __attribute__((ext_vector_type(8)));

// f32x2 -> packed bf16x2 (low = first arg).
__device__ __forceinline__ unsigned int pack_bf16x2(float a, float b) {
#if defined(__has_builtin) && __has_builtin(__builtin_amdgcn_cvt_pk_bf16_f32)
  union { v2bf v; unsigned int u; } c;
  c.v = __builtin_amdgcn_cvt_pk_bf16_f32(a, b);
  return c.u;
#else
  union { v2bf v; unsigned int u; } c;     // native bf16 fptrunc (1 cvt each)
  c.v[0] = (__bf16)a;
  c.v[1] = (__bf16)b;
  return c.u;
#endif
}

__device__ __forceinline__ v16bf cat8(v8bf lo, v8bf hi) {
  return __builtin_shufflevector(lo, hi, 0,1,2,3,4,5,6,7,8,9,10,11,12,13,14,15);
}
__device__ __forceinline__ v32bf cat16(v8bf a, v8bf b, v8bf c, v8bf d) {
  v16bf lo = cat8(a, b), hi = cat8(c, d);
  return __builtin_shufflevector(lo, hi,
      0,1,2,3,4,5,6,7,8,9,10,11,12,13,14,15,
      16,17,18,19,20,21,22,23,24,25,26,27,28,29,30,31);
}
__device__ __forceinline__ float sel4(float4 v, int p) {
  return p == 0 ? v.x : (p == 1 ? v.y : (p == 2 ? v.z : v.w));
}
// 2-of-4 selector from a mask float4: code = p0 | (p1<<2), p0 < p1.
__device__ __forceinline__ unsigned int mask_code(float4 m4) {
  unsigned int bits = (m4.x != 0.f ? 1u : 0u) | (m4.y != 0.f ? 2u : 0u) |
                      (m4.z != 0.f ? 4u : 0u) | (m4.w != 0.f ? 8u : 0u);
  if (bits == 0u) bits = 3u;
  const unsigned int p0 = __builtin_ctz(bits);
  const unsigned int b2 = bits & (bits - 1u);
  const unsigned int p1 = b2 ? __builtin_ctz(b2) : 3u;
  return p0 | (p1 << 2);
}

// Workspace: compressed W (1024 rows x 256 packed uints) + 32 index words
#define WS_W_U32   (1024 * 256)
#define WS_IDX_OFF WS_W_U32
#define WS_NEEDED  ((size_t)(WS_W_U32 + 32) * 4)

#define KSTEP 64
#define XPITCH 36   // uints per X row: 64 bf16 + 8 pad  (144B, 16B multiple)
#define WPITCH 20   // uints per Wc row: 32 bf16 + 8 pad (80B, 16B multiple)
#define OPITCH 68   // floats per out-stage row (272B, 16B multiple)

// ============ one-shot W compression (runs once per launch) ================
__global__ __launch_bounds__(256)
void ccl_compress(const float* __restrict__ W, const float* __restrict__ Mk,
                  unsigned int* __restrict__ ws, int K)
{
  const int r = blockIdx.x;        // out-channel row
  const int G = threadIdx.x;       // K-group 0..255
  const unsigned int code = mask_code(*(const float4*)(Mk + G * 4));
  const int p0 = code & 3, p1 = (code >> 2) & 3;
  const float4 w4 = *(const float4*)(W + (long)r * K + G * 4);
  ws[r * 256 + G] = pack_bf16x2(sel4(w4, p0), sel4(w4, p1));

  if (r == 0 && G < 32) {          // word t: groups t*8..t*8+7
    unsigned int word = 0;
#pragma unroll
    for (int j = 0; j < 8; ++j)
      word |= mask_code(*(const float4*)(Mk + G * 32 + j * 4)) << (j * 4);
    ws[WS_IDX_OFF + G] = word;
  }
}

// ============== pipelined 2:4 sparse SWMMAC GEMM (uses ws) =================
// Block tile: 128 out-channels (A) x 128 tokens (B), K step = 64 expanded.
// Double-buffered LDS: W(next) via async-to-LDS DMA, X(next) via regs, both
// overlapped with the current step's swmmacs; one barrier per step.
#define OFF_LX0 0
#define OFF_LX1 18432
#define OFF_LW0 36864                 // 2*128*36*4
#define OFF_LW1 47104
#define OFF_IDX 57344                 // + 2*128*20*4
#define SMEM_BYTES 57472              // also covers 34816B epilogue stage

__global__ __launch_bounds__(256)
void ccl_swmmac_pipe(const float* __restrict__ X, float* __restrict__ O,
                     const unsigned int* __restrict__ ws, int M, int N, int K)
{
  // Single shared block => LDS allocation starts at byte offset 0, so these
  // offsets are also the VDST addresses for async-to-LDS copies.
  __shared__ __align__(16) unsigned char smem[SMEM_BYTES];
  unsigned int* idxW = (unsigned int*)(smem + OFF_IDX);
  float*        lOut = (float*)smem;

  const int tid  = threadIdx.x;
  const int wid  = tid >> 5;
  const int lane = tid & 31;
  const int half = lane >> 4;
  const int mrow = lane & 15;

  const long outBase = (long)blockIdx.x * 128;
  const long tokBase = (long)blockIdx.y * 128;
  const int  waveOut = (wid & 3) << 5;
  const int  waveTok = (wid >> 2) << 6;

  if (tid < 32) idxW[tid] = ws[WS_IDX_OFF + tid];

  // --- staging helpers -----------------------------------------------------
  const int wrow = tid >> 2;              // W chunk mapping (2 chunks/thread)
  const int wq   = tid & 3;
  auto stage_w = [&](int k0, int bufOff) {
#pragma unroll
    for (int i = 0; i < 2; ++i) {
      const int row = wrow + (i << 6);    // c = tid + i*256 -> row += 64
      const unsigned lds_addr = (unsigned)(bufOff + row * (WPITCH * 4) + wq * 16);
      const unsigned long long ga = (unsigned long long)(const void*)
          (ws + (outBase + row) * 256 + (k0 >> 2) + wq * 4);
      asm volatile("global_load_async_to_lds_b128 %0, %1, off"
                   :: "v"(lds_addr), "v"(ga) : "memory");
    }
  };
  float4 xr[8];
  auto load_x = [&](int k0) {
#pragma unroll
    for (int i = 0; i < 8; ++i) {
      const int idx = tid + (i << 8);
      const int row = idx >> 4, q = idx & 15;
      xr[i] = *(const float4*)(X + (tokBase + row) * (long)K + k0 + q * 4);
    }
  };
  auto store_x = [&](unsigned int* lx) {
#pragma unroll
    for (int i = 0; i < 8; ++i) {
      const int idx = tid + (i << 8);
      const int row = idx >> 4, q = idx & 15;
      lx[row * XPITCH + q * 2    ] = pack_bf16x2(xr[i].x, xr[i].y);
      lx[row * XPITCH + q * 2 + 1] = pack_bf16x2(xr[i].z, xr[i].w);
    }
  };

  v8f acc[2][4];
#pragma unroll
  for (int i = 0; i < 2; ++i)
#pragma unroll
    for (int j = 0; j < 4; ++j)
#pragma unroll
      for (int e = 0; e < 8; ++e) acc[i][j][e] = 0.0f;

  // --- prologue: stage step 0 into buffer 0 --------------------------------
  stage_w(0, OFF_LW0);
  load_x(0);
  store_x((unsigned int*)(smem + OFF_LX0));
  asm volatile("s_wait_asynccnt 0" ::: "memory");
  __syncthreads();

  const int NSTEP = K / KSTEP;
  for (int s = 0; s < NSTEP; ++s) {
    const int cur = s & 1;
    const bool pf = (s + 1 < NSTEP);
    unsigned int* lXc = (unsigned int*)(smem + (cur ? OFF_LX1 : OFF_LX0));
    unsigned int* lWc = (unsigned int*)(smem + (cur ? OFF_LW1 : OFF_LW0));
    unsigned int* lXn = (unsigned int*)(smem + (cur ? OFF_LX0 : OFF_LX1));

    if (pf) {
      stage_w((s + 1) * KSTEP, cur ? OFF_LW0 : OFF_LW1);  // async DMA -> next
      load_x((s + 1) * KSTEP);                            // global -> regs
    }

    // Sparse index word: lanes 0-15 -> expanded K 0..31, lanes 16-31 -> 32..63
    const int iw = (int)idxW[s * 2 + half];

    // A fragments (compressed 16x32): half 0 -> c {0..7,16..23}, half 1 -> +8
    v16bf afrag[2];
#pragma unroll
    for (int ta = 0; ta < 2; ++ta) {
      const unsigned int* base = &lWc[(waveOut + ta * 16 + mrow) * WPITCH];
      afrag[ta] = cat8(*(const v8bf*)(base + half * 4),
                       *(const v8bf*)(base + half * 4 + 8));
    }
    // B fragments (64x16 col-major): half 0 -> K {0..15,32..47}, half 1 -> +16
#pragma unroll
    for (int tb = 0; tb < 4; ++tb) {
      const unsigned int* base = &lXc[(waveTok + tb * 16 + mrow) * XPITCH];
      const v32bf bfrag = cat16(*(const v8bf*)(base + half * 8),
                                *(const v8bf*)(base + half * 8 + 4),
                                *(const v8bf*)(base + half * 8 + 16),
                                *(const v8bf*)(base + half * 8 + 20));
#pragma unroll
      for (int ta = 0; ta < 2; ++ta)
        acc[ta][tb] = __builtin_amdgcn_swmmac_f32_16x16x64_bf16(
            false, afrag[ta], false, bfrag, acc[ta][tb], iw, false, false);
    }

    if (pf) store_x(lXn);                                 // regs -> next LDS
    asm volatile("s_wait_asynccnt 0" ::: "memory");
    __syncthreads();
  }

  // --- epilogue: D is [out][token]; transpose via LDS in 2 phases ----------
#pragma unroll
  for (int p = 0; p < 2; ++p) {
    if (((wid & 3) >> 1) == p) {
      const int outLoc = (wid & 1) << 5;
#pragma unroll
      for (int ta = 0; ta < 2; ++ta)
#pragma unroll
        for (int tb = 0; tb < 4; ++tb) {
          const int trow = waveTok + tb * 16 + mrow;
#pragma unroll
          for (int r = 0; r < 8; ++r)
            lOut[trow * OPITCH + outLoc + ta * 16 + r + half * 8] =
                acc[ta][tb][r];
        }
    }
    __syncthreads();
#pragma unroll
    for (int i = 0; i < 8; ++i) {
      const int idx = tid + (i << 8);
      const int row = idx >> 4;
      const int c4  = idx & 15;
      *(float4*)(O + (tokBase + row) * (long)N + outBase + p * 64 + c4 * 4) =
          *(const float4*)(lOut + row * OPITCH + c4 * 4);
    }
    __syncthreads();
  }
}

// ======== self-contained fallback (no workspace): compress in-loop =========
#define BOFF_LX  0
#define BOFF_LW  18432
#define BOFF_IDX 28672
#define BSMEM    34816

__global__ __launch_bounds__(256)
void ccl_swmmac_basic(const float* __restrict__ X, const float* __restrict__ W,
                      const float* __restrict__ Mk, float* __restrict__ O,
                      int M, int N, int K)
{
  __shared__ __align__(16) unsigned char smem[BSMEM];
  unsigned int* lX   = (unsigned int*)(smem + BOFF_LX);
  unsigned int* lW   = (unsigned int*)(smem + BOFF_LW);
  unsigned int* idxW = (unsigned int*)(smem + BOFF_IDX);
  float*        lOut = (float*)smem;

  const int tid  = threadIdx.x;
  const int wid  = tid >> 5;
  const int lane = tid & 31;
  const int half = lane >> 4;
  const int mrow = lane & 15;

  const long outBase = (long)blockIdx.x * 128;
  const long tokBase = (long)blockIdx.y * 128;
  const int  waveOut = (wid & 3) << 5;
  const int  waveTok = (wid >> 2) << 6;

  if (tid < 32) {
    unsigned int word = 0;
#pragma unroll
    for (int j = 0; j < 8; ++j)
      word |= mask_code(*(const float4*)(Mk + tid * 32 + j * 4)) << (j * 4);
    idxW[tid] = word;
  }
  __syncthreads();

  v8f acc[2][4];
#pragma unroll
  for (int i = 0; i < 2; ++i)
#pragma unroll
    for (int j = 0; j < 4; ++j)
#pragma unroll
      for (int e = 0; e < 8; ++e) acc[i][j][e] = 0.0f;

  for (int k0 = 0; k0 < K; k0 += KSTEP) {
#pragma unroll
    for (int i = 0; i < 8; ++i) {
      const int idx = tid + (i << 8);
      const int row = idx >> 4;
      const int g   = idx & 15;
      const int G   = (k0 >> 2) + g;
      const unsigned int code = (idxW[G >> 3] >> ((G & 7) * 4)) & 0xFu;
      const int p0 = code & 3, p1 = (code >> 2) & 3;
      const float4 w4 =
          *(const float4*)(W + (outBase + row) * (long)K + k0 + g * 4);
      lW[row * WPITCH + g] = pack_bf16x2(sel4(w4, p0), sel4(w4, p1));
    }
#pragma unroll
    for (int i = 0; i < 8; ++i) {
      const int idx = tid + (i << 8);
      const int row = idx >> 4;
      const int q   = idx & 15;
      const float4 xv =
          *(const float4*)(X + (tokBase + row) * (long)K + k0 + q * 4);
      lX[row * XPITCH + q * 2    ] = pack_bf16x2(xv.x, xv.y);
      lX[row * XPITCH + q * 2 + 1] = pack_bf16x2(xv.z, xv.w);
    }
    __syncthreads();

    const int iw = (int)idxW[(k0 >> 5) + half];
    v16bf afrag[2];
#pragma unroll
    for (int ta = 0; ta < 2; ++ta) {
      const unsigned int* base = &lW[(waveOut + ta * 16 + mrow) * WPITCH];
      afrag[ta] = cat8(*(const v8bf*)(base + half * 4),
                       *(const v8bf*)(base + half * 4 + 8));
    }
#pragma unroll
    for (int tb = 0; tb < 4; ++tb) {
      const unsigned int* base = &lX[(waveTok + tb * 16 + mrow) * XPITCH];
      const v32bf bfrag = cat16(*(const v8bf*)(base + half * 8),
                                *(const v8bf*)(base + half * 8 + 4),
                                *(const v8bf*)(base + half * 8 + 16),
                                *(const v8bf*)(base + half * 8 + 20));
#pragma unroll
      for (int ta = 0; ta < 2; ++ta)
        acc[ta][tb] = __builtin_amdgcn_swmmac_f32_16x16x64_bf16(
            false, afrag[ta], false, bfrag, acc[ta][tb], iw, false, false);
    }
    __syncthreads();
  }

#pragma unroll
  for (int p = 0; p < 2; ++p) {
    if (((wid & 3) >> 1) == p) {
      const int outLoc = (wid & 1) << 5;
#pragma unroll
      for (int ta = 0; ta < 2; ++ta)
#pragma unroll
        for (int tb = 0; tb < 4; ++tb) {
          const int trow = waveTok + tb * 16 + mrow;
#pragma unroll
          for (int r = 0; r < 8; ++r)
            lOut[trow * OPITCH + outLoc + ta * 16 + r + half * 8] =
                acc[ta][tb][r];
        }
    }
    __syncthreads();
#pragma unroll
    for (int i = 0; i < 8; ++i) {
      const int idx = tid + (i << 8);
      const int row = idx >> 4;
      const int c4  = idx & 15;
      *(float4*)(O + (tokBase + row) * (long)N + outBase + p * 64 + c4 * 4) =
          *(const float4*)(lOut + row * OPITCH + c4 * 4);
    }
    __syncthreads();
  }
}

extern "C" void kernel_launch(void* const* d_in, const int* in_sizes, int n_in,
                              void* d_out, int out_size, void* d_ws, size_t ws_size,
                              hipStream_t stream) {
  (void)n_in; (void)out_size;
  const float* x  = (const float*)d_in[0];
  const float* w  = (const float*)d_in[1];
  const float* mk = (const float*)d_in[2];
  float* o = (float*)d_out;

  const int K = 1024, N = 1024;
  const int M = in_sizes[0] / K;   // 32768

  dim3 grid((unsigned)(N / 128), (unsigned)(M / 128));   // 8 x 256
  if (ws_size >= WS_NEEDED && d_ws != nullptr) {
    unsigned int* ws = (unsigned int*)d_ws;
    ccl_compress<<<dim3(1024), 256, 0, stream>>>(w, mk, ws, K);
    ccl_swmmac_pipe<<<grid, 256, 0, stream>>>(x, o, ws, M, N, K);
  } else {
    ccl_swmmac_basic<<<grid, 256, 0, stream>>>(x, w, mk, o, M, N, K);
  }
}